// ConvNetWithMambaAndSelfAttention_82171314307329
// MI455X (gfx1250) — compile-verified
//
#include <hip/hip_runtime.h>
#include <math.h>

// ---------------------------------------------------------------------------
// CDNA5 (gfx1250, wave32) forward pass for ConvNet+Mamba+SelfAttention.
// All linear layers -> V_WMMA_F32_16X16X4_F32 (fp32 WMMA, fp32 accumulate).
// Convs / BN / S6 scan -> VALU kernels (tiny, latency-bound workload).
// ---------------------------------------------------------------------------

typedef __attribute__((ext_vector_type(2))) float v2f;
typedef __attribute__((ext_vector_type(8))) float v8f;

static __device__ __forceinline__ float act_apply(float v, int act) {
  if (act == 1) return fmaxf(v, 0.f);                                  // relu
  if (act == 2) return v / (1.f + __expf(-v));                         // silu
  if (act == 3) return (v > 20.f) ? v : log1pf(__expf(v));             // softplus
  return v;
}

// ---------------------------------------------------------------------------
// GEMM: Y[M,N](stride ldy) = act( X[M,K] @ W[N,K]^T + bias[N] )
// One wave computes one 16x16 tile via V_WMMA_F32_16X16X4_F32.
// A frag (16x4): lanes 0-15 K=0,1 ; lanes 16-31 K=2,3 (row = lane%16).
// B frag (4x16): same split on K, col = lane%16 (W is row-major (N,K)).
// D frag: VGPR r -> M = 16*tm + r (+8 for lanes 16-31), N = 16*tn + lane%16.
// Requires: M%16==0, N%16==0, K%4==0 (true for every linear in this net).
// ---------------------------------------------------------------------------
__global__ void gemm_wmma_kernel(const float* __restrict__ X, const float* __restrict__ W,
                                 const float* __restrict__ bias, float* __restrict__ Y,
                                 int M, int N, int K, int act, int ldy) {
  int wave = (int)((blockIdx.x * blockDim.x + threadIdx.x) >> 5);
  int lane = (int)(threadIdx.x & 31);
  int nt = N >> 4;
  if (wave >= (M >> 4) * nt) return;
  int tm = wave / nt, tn = wave - tm * nt;
  int mrow = (tm << 4) + (lane & 15);
  int ncol = (tn << 4) + (lane & 15);
  int kh = (lane >> 4) << 1;  // 0 or 2
  const float* Xp = X + (size_t)mrow * K + kh;
  const float* Wp = W + (size_t)ncol * K + kh;
  v8f c = {};
  for (int k = 0; k < K; k += 4) {
    v2f a = *(const v2f*)(Xp + k);
    v2f b = *(const v2f*)(Wp + k);
    c = __builtin_amdgcn_wmma_f32_16x16x4_f32(false, a, false, b, (short)0, c, false, false);
  }
  float bv = bias[ncol];
  int mb = (tm << 4) + ((lane >> 4) << 3);
#pragma unroll
  for (int r = 0; r < 8; ++r) {
    Y[(size_t)(mb + r) * ldy + ncol] = act_apply(c[r] + bv, act);
  }
}

// ---------------------------------------------------------------------------
// Direct conv2d (VALID) + maxpool(P) + relu.  out layout (B,Co,Hp,Wp).
// ---------------------------------------------------------------------------
__global__ void conv2d_pool_relu_kernel(const float* __restrict__ in, const float* __restrict__ w,
                                        const float* __restrict__ bias, float* __restrict__ out,
                                        int B, int Ci, int H, int Wd, int Co, int KH, int KW, int P) {
  int Ho = H - KH + 1, Wo = Wd - KW + 1;
  int Hp = Ho / P, Wp = Wo / P;
  int total = B * Co * Hp * Wp;
  int idx = blockIdx.x * blockDim.x + threadIdx.x;
  if (idx >= total) return;
  int px = idx % Wp; int t = idx / Wp;
  int py = t % Hp;   t /= Hp;
  int co = t % Co;   int b = t / Co;
  float m = -3.4e38f;
  for (int dy = 0; dy < P; ++dy)
    for (int dx = 0; dx < P; ++dx) {
      int oy = py * P + dy, ox = px * P + dx;
      float acc = bias[co];
      for (int ci = 0; ci < Ci; ++ci) {
        const float* ip = in + ((size_t)(b * Ci + ci) * H + oy) * Wd + ox;
        const float* wp = w + (size_t)(co * Ci + ci) * KH * KW;
        for (int ky = 0; ky < KH; ++ky)
          for (int kx = 0; kx < KW; ++kx)
            acc += ip[ky * Wd + kx] * wp[ky * KW + kx];
      }
      m = fmaxf(m, acc);
    }
  out[idx] = fmaxf(m, 0.f);
}

// ---------------------------------------------------------------------------
// Direct conv1d (pad) + maxpool(P) + act.  Used for the 1D branch (relu) and
// the mamba channel-mixing conv (Ci=Co=45 over the 128-wide feature axis, silu).
// ---------------------------------------------------------------------------
__global__ void conv1d_pool_act_kernel(const float* __restrict__ in, const float* __restrict__ w,
                                       const float* __restrict__ bias, float* __restrict__ out,
                                       int B, int Ci, int L, int Co, int Kk, int pad, int P, int act) {
  int Lc = L + 2 * pad - Kk + 1;
  int Lp = Lc / P;
  int total = B * Co * Lp;
  int idx = blockIdx.x * blockDim.x + threadIdx.x;
  if (idx >= total) return;
  int lp = idx % Lp; int t = idx / Lp;
  int co = t % Co;   int b = t / Co;
  float m = -3.4e38f;
  for (int dp = 0; dp < P; ++dp) {
    int lc = lp * P + dp;
    float acc = bias[co];
    for (int ci = 0; ci < Ci; ++ci) {
      const float* ip = in + (size_t)(b * Ci + ci) * L;
      const float* wp = w + (size_t)(co * Ci + ci) * Kk;
      for (int kk = 0; kk < Kk; ++kk) {
        int pos = lc + kk - pad;
        if (pos >= 0 && pos < L) acc += ip[pos] * wp[kk];
      }
    }
    m = fmaxf(m, acc);
  }
  out[idx] = act_apply(m, act);
}

// ---------------------------------------------------------------------------
// Training-mode BatchNorm over (N,*,S) per channel, in-place.  One block/chan.
// ---------------------------------------------------------------------------
__global__ void bn_kernel(float* __restrict__ x, const float* __restrict__ g,
                          const float* __restrict__ bta, int Bn, int C, int S) {
  int c = blockIdx.x;
  int tid = threadIdx.x;
  int cnt = Bn * S;
  float s = 0.f, s2 = 0.f;
  for (int i = tid; i < cnt; i += blockDim.x) {
    int n = i / S, sp = i - n * S;
    float v = x[(size_t)(n * C + c) * S + sp];
    s += v; s2 += v * v;
  }
  __shared__ float sh1[256], sh2[256];
  sh1[tid] = s; sh2[tid] = s2;
  __syncthreads();
  for (int st = 128; st > 0; st >>= 1) {
    if (tid < st) { sh1[tid] += sh1[tid + st]; sh2[tid] += sh2[tid + st]; }
    __syncthreads();
  }
  float mean = sh1[0] / (float)cnt;
  float var  = sh2[0] / (float)cnt - mean * mean;
  float sc = rsqrtf(var + 1e-5f) * g[c];
  float sb = bta[c] - mean * sc;
  for (int i = tid; i < cnt; i += blockDim.x) {
    int n = i / S, sp = i - n * S;
    float* p = &x[(size_t)(n * C + c) * S + sp];
    *p = *p * sc + sb;
  }
}

// (B,C,L) -> (B,L,C)
__global__ void transpose_bcl_kernel(const float* __restrict__ in, float* __restrict__ out,
                                     int B, int C, int L) {
  int total = B * C * L;
  int idx = blockIdx.x * blockDim.x + threadIdx.x;
  if (idx >= total) return;
  int l = idx % L; int t = idx / L;
  int c = t % C;   int b = t / C;
  out[((size_t)b * L + l) * C + c] = in[idx];
}

// RMSNorm over last dim 64: one wave per row, 2 elems/lane, shfl reduction.
__global__ void rmsnorm_kernel(const float* __restrict__ x, const float* __restrict__ w,
                               float* __restrict__ y, int rows) {
  int wave = (int)((blockIdx.x * blockDim.x + threadIdx.x) >> 5);
  int lane = (int)(threadIdx.x & 31);
  if (wave >= rows) return;
  const float* xr = x + (size_t)wave * 64;
  float v0 = xr[lane], v1 = xr[lane + 32];
  float ss = v0 * v0 + v1 * v1;
  for (int off = 16; off > 0; off >>= 1) ss += __shfl_xor(ss, off, 32);
  float inv = rsqrtf(ss * (1.f / 64.f) + 1e-5f);
  y[(size_t)wave * 64 + lane]      = v0 * inv * w[lane];
  y[(size_t)wave * 64 + lane + 32] = v1 * inv * w[lane + 32];
}

// ---------------------------------------------------------------------------
// S6 selective scan.  One wave per (b,d): N=128 state held 4/lane in regs;
// per step: h = exp(delta*A)*h + delta*x*B ; y = sum_n C*h via wave reduce.
// Writes xs = silu(y).
// ---------------------------------------------------------------------------
__global__ void s6_scan_kernel(const float* __restrict__ dlt, const float* __restrict__ Bm,
                               const float* __restrict__ Cm, const float* __restrict__ xin,
                               const float* __restrict__ A, float* __restrict__ xs,
                               int Bn, int L, int D, int N) {
  int wave = (int)((blockIdx.x * blockDim.x + threadIdx.x) >> 5);
  int lane = (int)(threadIdx.x & 31);
  if (wave >= Bn * D) return;
  int b = wave / D, d = wave - b * D;
  const float* Ad = A + (size_t)d * N;
  float a0 = Ad[lane], a1 = Ad[lane + 32], a2 = Ad[lane + 64], a3 = Ad[lane + 96];
  float h0 = 0.f, h1 = 0.f, h2 = 0.f, h3 = 0.f;
  for (int t = 0; t < L; ++t) {
    size_t rb = (size_t)(b * L + t);
    float delta = dlt[rb * D + d];
    float xv    = xin[rb * D + d];
    float sb = delta * xv;
    const float* Bp = Bm + rb * N;
    const float* Cp = Cm + rb * N;
    h0 = __expf(delta * a0) * h0 + sb * Bp[lane];
    h1 = __expf(delta * a1) * h1 + sb * Bp[lane + 32];
    h2 = __expf(delta * a2) * h2 + sb * Bp[lane + 64];
    h3 = __expf(delta * a3) * h3 + sb * Bp[lane + 96];
    float acc = Cp[lane] * h0 + Cp[lane + 32] * h1 + Cp[lane + 64] * h2 + Cp[lane + 96] * h3;
    for (int off = 16; off > 0; off >>= 1) acc += __shfl_xor(acc, off, 32);
    if (lane == 0) xs[rb * D + d] = acc / (1.f + __expf(-acc));  // fused silu
  }
}

__global__ void mul_inplace_kernel(float* __restrict__ a, const float* __restrict__ b, int n) {
  int i = blockIdx.x * blockDim.x + threadIdx.x;
  if (i < n) a[i] *= b[i];
}

__global__ void copy2d_kernel(const float* __restrict__ src, int ss, float* __restrict__ dst,
                              int ds, int rows, int cols) {
  int total = rows * cols;
  int idx = blockIdx.x * blockDim.x + threadIdx.x;
  if (idx >= total) return;
  int r = idx / cols, c = idx - r * cols;
  dst[(size_t)r * ds + c] = src[(size_t)r * ss + c];
}

// classifier (32x64)@(8x64)^T + softmax; writes logits and probas.
__global__ void cls_softmax_kernel(const float* __restrict__ x, const float* __restrict__ w,
                                   const float* __restrict__ bias, float* __restrict__ logits,
                                   float* __restrict__ probas) {
  int row = threadIdx.x;
  if (row >= 32) return;
  float lv[8];
  float mx = -3.4e38f;
  const float* xr = x + (size_t)row * 64;
  for (int o = 0; o < 8; ++o) {
    float acc = bias[o];
    const float* wr = w + (size_t)o * 64;
    for (int k = 0; k < 64; ++k) acc += xr[k] * wr[k];
    lv[o] = acc;
    mx = fmaxf(mx, acc);
  }
  float e[8], s = 0.f;
  for (int o = 0; o < 8; ++o) { e[o] = __expf(lv[o] - mx); s += e[o]; }
  for (int o = 0; o < 8; ++o) {
    logits[row * 8 + o] = lv[o];
    probas[row * 8 + o] = e[o] / s;
  }
}

// ---------------------------------------------------------------------------
extern "C" void kernel_launch(void* const* d_in, const int* in_sizes, int n_in,
                              void* d_out, int out_size, void* d_ws, size_t ws_size,
                              hipStream_t stream) {
  (void)in_sizes; (void)n_in; (void)out_size; (void)ws_size;

  // ---- inputs (setup_inputs() dict insertion order, params recursed in order)
  int t = 0;
  auto nxt = [&]() { return (const float*)d_in[t++]; };
  const float* x1 = nxt();                 // (32,3,60,60)
  const float* x2 = nxt();                 // (32,1,433)
  const float *c2d1_w = nxt(), *c2d1_b = nxt();
  const float *c2d2_w = nxt(), *c2d2_b = nxt();
  const float *c2d3_w = nxt(), *c2d3_b = nxt();
  const float *c2d4_w = nxt(), *c2d4_b = nxt();
  const float *bn2d1_g = nxt(), *bn2d1_b = nxt();
  const float *bn2d2_g = nxt(), *bn2d2_b = nxt();
  const float *bn2d3_g = nxt(), *bn2d3_b = nxt();
  const float *bn2d4_g = nxt(), *bn2d4_b = nxt();
  const float *att2_w1 = nxt(), *att2_b1 = nxt();
  const float *att2_w2 = nxt(), *att2_b2 = nxt();
  const float *c1d1_w = nxt(), *c1d1_b = nxt();
  const float *c1d2_w = nxt(), *c1d2_b = nxt();
  const float *c1d3_w = nxt(), *c1d3_b = nxt();
  const float *bn1d1_g = nxt(), *bn1d1_b = nxt();
  const float *bn1d2_g = nxt(), *bn1d2_b = nxt();
  const float *bn1d3_g = nxt(), *bn1d3_b = nxt();
  const float *att1_w1 = nxt(), *att1_b1 = nxt();
  const float *att1_w2 = nxt(), *att1_b2 = nxt();
  struct MB {
    const float *norm_w, *inp_w, *inp_b, *conv_w, *conv_b, *cl_w, *cl_b;
    const float *d_w, *d_b, *out_w, *out_b;
    const float *fc1_w, *fc1_b, *fc2_w, *fc2_b, *fc3_w, *fc3_b, *A;
  } mb[3];
  for (int i = 0; i < 3; ++i) {
    mb[i].norm_w = nxt();
    mb[i].inp_w = nxt();  mb[i].inp_b = nxt();
    mb[i].conv_w = nxt(); mb[i].conv_b = nxt();
    mb[i].cl_w = nxt();   mb[i].cl_b = nxt();
    mb[i].d_w = nxt();    mb[i].d_b = nxt();
    mb[i].out_w = nxt();  mb[i].out_b = nxt();
    mb[i].fc1_w = nxt();  mb[i].fc1_b = nxt();
    mb[i].fc2_w = nxt();  mb[i].fc2_b = nxt();
    mb[i].fc3_w = nxt();  mb[i].fc3_b = nxt();
    mb[i].A = nxt();
  }
  const float *f_w1 = nxt(), *f_b1 = nxt();
  const float *f_w2 = nxt(), *f_b2 = nxt();
  const float *f_w3 = nxt(), *f_b3 = nxt();
  const float *cls_w = nxt(), *cls_b = nxt();

  float* out = (float*)d_out;
  // d_out layout: [logits 256 | probas 256 | x1o 36864 | x2o 92160 | x 2048]
  float* out_logits = out;
  float* out_probas = out + 256;
  float* out_x1o    = out + 512;
  float* out_x2o    = out + 37376;
  float* out_x      = out + 129536;

  // ---- workspace carve-out (floats)
  float* ws = (float*)d_ws;
  size_t off = 0;
  auto alloc = [&](size_t n) { float* p = ws + off; off += n; return p; };
  float* p2a  = alloc(401408);  // (32,16,28,28)
  float* p2b  = alloc(147456);  // (32,32,12,12)
  float* p2c  = alloc(51200);   // (32,64,5,5)
  float* p2d  = alloc(36864);   // (32,128,3,3) = a (32,1152)
  float* h64  = alloc(2048);    // (32,64)
  float* cat  = alloc(129024);  // (32,4032)
  float* c1   = alloc(73216);   // (32,16,143)
  float* c2b  = alloc(48128);   // (32,32,47)
  float* c3   = alloc(92160);   // (32,64,45)
  float* seqA = alloc(92160);   // (1440,64)
  float* seqB = alloc(92160);
  float* mid32 = alloc(46080);  // (1440,32)
  float* xnb  = alloc(92160);   // (1440,64)
  float* xpb  = alloc(184320);  // (1440,128)
  float* xcv  = alloc(184320);
  float* xc2  = alloc(184320);
  float* dltb = alloc(184320);
  float* Bmb  = alloc(184320);
  float* Cmb  = alloc(184320);
  float* xsb  = alloc(184320);
  float* xrb  = alloc(184320);
  float* f1o  = alloc(8192);    // (32,256)
  float* f2o  = alloc(4096);    // (32,128)

  auto blocks = [](int n) { return (n + 255) / 256; };
  auto gemm = [&](const float* X, const float* W, const float* bias, float* Y,
                  int M, int N, int K, int act, int ldy) {
    int waves = (M / 16) * (N / 16);
    gemm_wmma_kernel<<<blocks(waves * 32), 256, 0, stream>>>(X, W, bias, Y, M, N, K, act, ldy);
  };

  // ================= 2D CNN branch =================
  conv2d_pool_relu_kernel<<<blocks(401408), 256, 0, stream>>>(x1, c2d1_w, c2d1_b, p2a, 32, 3, 60, 60, 16, 5, 5, 2);
  bn_kernel<<<16, 256, 0, stream>>>(p2a, bn2d1_g, bn2d1_b, 32, 16, 784);
  conv2d_pool_relu_kernel<<<blocks(147456), 256, 0, stream>>>(p2a, c2d2_w, c2d2_b, p2b, 32, 16, 28, 28, 32, 5, 5, 2);
  bn_kernel<<<32, 256, 0, stream>>>(p2b, bn2d2_g, bn2d2_b, 32, 32, 144);
  conv2d_pool_relu_kernel<<<blocks(51200), 256, 0, stream>>>(p2b, c2d3_w, c2d3_b, p2c, 32, 32, 12, 12, 64, 3, 3, 2);
  bn_kernel<<<64, 256, 0, stream>>>(p2c, bn2d3_g, bn2d3_b, 32, 64, 25);
  conv2d_pool_relu_kernel<<<blocks(36864), 256, 0, stream>>>(p2c, c2d4_w, c2d4_b, p2d, 32, 64, 5, 5, 128, 3, 3, 1);
  bn_kernel<<<128, 256, 0, stream>>>(p2d, bn2d4_g, bn2d4_b, 32, 128, 9);
  // att2 MLP: (32,1152)->relu(32,64)->(32,1152), result into cat[:, 0:1152]
  gemm(p2d, att2_w1, att2_b1, h64, 32, 64, 1152, 1, 64);
  gemm(h64, att2_w2, att2_b2, cat, 32, 1152, 64, 0, 4032);
  copy2d_kernel<<<blocks(36864), 256, 0, stream>>>(cat, 4032, out_x1o, 1152, 32, 1152);

  // ================= 1D CNN branch =================
  conv1d_pool_act_kernel<<<blocks(73216), 256, 0, stream>>>(x2, c1d1_w, c1d1_b, c1, 32, 1, 433, 16, 5, 0, 3, 1);
  bn_kernel<<<16, 256, 0, stream>>>(c1, bn1d1_g, bn1d1_b, 32, 16, 143);
  conv1d_pool_act_kernel<<<blocks(48128), 256, 0, stream>>>(c1, c1d2_w, c1d2_b, c2b, 32, 16, 143, 32, 3, 0, 3, 1);
  bn_kernel<<<32, 256, 0, stream>>>(c2b, bn1d2_g, bn1d2_b, 32, 32, 47);
  conv1d_pool_act_kernel<<<blocks(92160), 256, 0, stream>>>(c2b, c1d3_w, c1d3_b, c3, 32, 32, 47, 64, 3, 0, 1, 1);
  bn_kernel<<<64, 256, 0, stream>>>(c3, bn1d3_g, bn1d3_b, 32, 64, 45);
  transpose_bcl_kernel<<<blocks(92160), 256, 0, stream>>>(c3, seqA, 32, 64, 45);  // (32,45,64)
  // att1 MLP: (1440,64)->relu(1440,32)->(1440,64)
  gemm(seqA, att1_w1, att1_b1, mid32, 1440, 32, 64, 1, 32);
  gemm(mid32, att1_w2, att1_b2, seqB, 1440, 64, 32, 0, 64);

  // ================= Mamba blocks =================
  float* cur = seqB;
  float* nxtb = seqA;
  for (int i = 0; i < 3; ++i) {
    const MB& m = mb[i];
    rmsnorm_kernel<<<blocks(1440 * 32), 256, 0, stream>>>(cur, m.norm_w, xnb, 1440);
    gemm(xnb, m.inp_w, m.inp_b, xpb, 1440, 128, 64, 0, 128);                  // (1440,128)
    // channel-mix conv over L=45 channels, spatial=feature axis (128), pad 1, silu
    conv1d_pool_act_kernel<<<blocks(184320), 256, 0, stream>>>(xpb, m.conv_w, m.conv_b, xcv, 32, 45, 128, 45, 3, 1, 1, 2);
    gemm(xcv, m.cl_w, m.cl_b, xc2, 1440, 128, 128, 0, 128);                   // s6 input x
    gemm(xc2, m.fc1_w, m.fc1_b, dltb, 1440, 128, 128, 3, 128);                // softplus -> delta
    gemm(xc2, m.fc2_w, m.fc2_b, Bmb, 1440, 128, 128, 0, 128);
    gemm(xc2, m.fc3_w, m.fc3_b, Cmb, 1440, 128, 128, 0, 128);
    s6_scan_kernel<<<blocks(32 * 128 * 32), 256, 0, stream>>>(dltb, Bmb, Cmb, xc2, m.A, xsb, 32, 45, 128, 128);
    gemm(xnb, m.d_w, m.d_b, xrb, 1440, 128, 64, 2, 128);                      // silu gate
    mul_inplace_kernel<<<blocks(184320), 256, 0, stream>>>(xsb, xrb, 184320);
    gemm(xsb, m.out_w, m.out_b, nxtb, 1440, 64, 128, 0, 64);
    float* tmp = cur; cur = nxtb; nxtb = tmp;
  }
  // cur = final sequence (32,45,64) = x2o (32,2880)
  copy2d_kernel<<<blocks(92160), 256, 0, stream>>>(cur, 2880, cat + 1152, 4032, 32, 2880);
  copy2d_kernel<<<blocks(92160), 256, 0, stream>>>(cur, 92160, out_x2o, 92160, 1, 92160);

  // ================= Head =================
  gemm(cat, f_w1, f_b1, f1o, 32, 256, 4032, 1, 256);
  gemm(f1o, f_w2, f_b2, f2o, 32, 128, 256, 1, 128);
  gemm(f2o, f_w3, f_b3, out_x, 32, 64, 128, 0, 64);
  cls_softmax_kernel<<<1, 32, 0, stream>>>(out_x, cls_w, cls_b, out_logits, out_probas);
}